// KNNTopoLoss_88338887344887
// MI455X (gfx1250) — compile-verified
//
#include <hip/hip_runtime.h>
#include <hip/hip_bf16.h>
#include <float.h>

typedef __attribute__((ext_vector_type(2))) float v2f;
typedef __attribute__((ext_vector_type(4))) float v4f;
typedef __attribute__((ext_vector_type(8))) float v8f;

#define N_PTS   8192
#define KNN     8
#define ROWS    32      // output rows per workgroup
#define PANEL   32      // columns per panel
#define THREADS 128     // 4 wave32

// ---------------------------------------------------------------------------
// Kernel 1: per-row squared norms
// ---------------------------------------------------------------------------
__global__ void row_norms_kernel(const float* __restrict__ x, int dim,
                                 float* __restrict__ sq) {
  int row = blockIdx.x * blockDim.x + threadIdx.x;
  if (row >= N_PTS) return;
  const float* p = x + (size_t)row * dim;
  float s = 0.f;
  for (int k = 0; k < dim; ++k) { float v = p[k]; s += v * v; }
  sq[row] = s;
}

// ---------------------------------------------------------------------------
// Kernel 2: fused fp32-WMMA Gram + squared-distance + per-row top-8 indices.
// 128 threads (4 waves) per block, 32 rows per block. B panels are streamed
// into LDS with double-buffered global_load_async_to_lds_b128 (ASYNCcnt),
// overlapping the next panel's copy with the current panel's WMMA work.
// Wave w computes the 16x16 tile (rowgrp = w>>1, colgrp = w&1).
// ---------------------------------------------------------------------------
__launch_bounds__(THREADS)
__global__ void knn_topk_kernel(const float* __restrict__ x, int dim,
                                const float* __restrict__ sq,
                                int* __restrict__ topk) {
  __shared__ float As[ROWS][132];        // 16.9 KB, padded stride (16B-aligned rows)
  __shared__ float Bs[2][PANEL][132];    // 33.8 KB, double-buffered async dest
  __shared__ float dtile[ROWS][PANEL];   // 4 KB distance tile (reused as cand-d)
  __shared__ int   ci[ROWS][PANEL];      // 4 KB candidate indices

  const int tid  = threadIdx.x;
  const int lane = tid & 31;
  const int wave = tid >> 5;
  const int rg   = wave >> 1;            // tile row group 0..1
  const int cg   = wave & 1;             // tile col group 0..1
  const int row0 = blockIdx.x * ROWS;

  // Stage A block (ROWS x dim) with float4 loads.
  {
    const int nf4 = ROWS * dim / 4;
    for (int f = tid; f < nf4; f += THREADS) {
      int e = f * 4;
      int r = e / dim, k = e % dim;
      *(v4f*)&As[r][k] = *(const v4f*)&x[(size_t)(row0 + r) * dim + k];
    }
  }

  // Per-thread partial top-8: 4 threads per row, strided columns.
  const int prow = tid >> 2;             // 0..31
  const int psub = tid & 3;              // 0..3
  float best[KNN];
  int   bidx[KNN];
#pragma unroll
  for (int i = 0; i < KNN; ++i) { best[i] = FLT_MAX; bidx[i] = -1; }

  const int nIssue = (PANEL * dim / 4) / THREADS;  // b128 chunks per thread

  // Prologue: async-stage panel 0 into buffer 0 (panel rows are contiguous
  // in memory, so the global side is a linear 16B stream from the panel base).
  {
    unsigned long long pan = (unsigned long long)(const void*)x;
    for (int i = 0; i < nIssue; ++i) {
      int f = tid + i * THREADS;         // float4 id within panel
      int e = f * 4;
      int r = e / dim, k = e % dim;
      unsigned lds  = (unsigned)(uintptr_t)&Bs[0][r][k];
      unsigned voff = (unsigned)(f * 16);
      asm volatile("global_load_async_to_lds_b128 %0, %1, %2"
                   :: "v"(lds), "v"(voff), "s"(pan) : "memory");
    }
  }

  const int NPAN = N_PTS / PANEL;
  for (int p = 0; p < NPAN; ++p) {
    const int buf = p & 1;
    const int j0  = p * PANEL;

    // Drain this panel's async copies, make them visible to all waves.
    asm volatile("s_wait_asynccnt 0x0" ::: "memory");
    __syncthreads();

    // Kick off next panel into the other buffer; overlaps with WMMA below.
    if (p + 1 < NPAN) {
      unsigned long long pan =
          (unsigned long long)(const void*)(x + (size_t)(j0 + PANEL) * dim);
      for (int i = 0; i < nIssue; ++i) {
        int f = tid + i * THREADS;
        int e = f * 4;
        int r = e / dim, k = e % dim;
        unsigned lds  = (unsigned)(uintptr_t)&Bs[buf ^ 1][r][k];
        unsigned voff = (unsigned)(f * 16);
        asm volatile("global_load_async_to_lds_b128 %0, %1, %2"
                     :: "v"(lds), "v"(voff), "s"(pan) : "memory");
      }
    }

    // fp32 WMMA K-loop: 16x16 Gram tile.
    v8f acc = {};
    const int mrow = lane & 15;
    const int koff = (lane >> 4) * 2;    // lanes 0-15: K=0,1 ; 16-31: K=2,3
    for (int k0 = 0; k0 < dim; k0 += 4) {
      v2f a, b;
      a.x = As[rg * 16 + mrow][k0 + koff];
      a.y = As[rg * 16 + mrow][k0 + koff + 1];
      b.x = Bs[buf][cg * 16 + mrow][k0 + koff];
      b.y = Bs[buf][cg * 16 + mrow][k0 + koff + 1];
      acc = __builtin_amdgcn_wmma_f32_16x16x4_f32(
          false, a, false, b, (short)0, acc, false, false);
    }

    // d = ||xi||^2 + ||xj||^2 - 2*dot ; diagonal -> +inf
    const int ncol = lane & 15;
    const int gcol = j0 + cg * 16 + ncol;
    const float sqc = sq[gcol];
#pragma unroll
    for (int v = 0; v < 8; ++v) {
      int m = rg * 16 + v + ((lane >> 4) << 3);
      int grow = row0 + m;
      float d = sq[grow] + sqc - 2.0f * acc[v];
      if (grow == gcol) d = FLT_MAX;
      dtile[m][cg * 16 + ncol] = d;
    }
    __syncthreads();

    // Partial top-8 update: each thread scans its 8 strided columns.
#pragma unroll
    for (int q = 0; q < 8; ++q) {
      int c = psub + (q << 2);
      float d = dtile[prow][c];
      if (d < best[KNN - 1]) {
        int gi = j0 + c;
        int pp = KNN - 1;
        while (pp > 0 && best[pp - 1] > d) {
          best[pp] = best[pp - 1]; bidx[pp] = bidx[pp - 1]; --pp;
        }
        best[pp] = d; bidx[pp] = gi;
      }
    }
  }

  __syncthreads();
  // Dump 4 partial lists per row (32 candidates); reuse dtile for distances.
#pragma unroll
  for (int i = 0; i < KNN; ++i) {
    dtile[prow][psub * KNN + i] = best[i];
    ci[prow][psub * KNN + i]    = bidx[i];
  }
  __syncthreads();

  // One thread per row merges 32 candidates -> final top-8 (index tiebreak).
  if (tid < ROWS) {
    float fb[KNN]; int fi[KNN];
#pragma unroll
    for (int i = 0; i < KNN; ++i) { fb[i] = FLT_MAX; fi[i] = 0x7fffffff; }
    for (int c = 0; c < PANEL; ++c) {
      float d = dtile[tid][c];
      int   g = ci[tid][c];
      if (g < 0) continue;
      if (d < fb[KNN - 1] || (d == fb[KNN - 1] && g < fi[KNN - 1])) {
        int pp = KNN - 1;
        while (pp > 0 && (fb[pp - 1] > d || (fb[pp - 1] == d && fi[pp - 1] > g))) {
          fb[pp] = fb[pp - 1]; fi[pp] = fi[pp - 1]; --pp;
        }
        fb[pp] = d; fi[pp] = g;
      }
    }
#pragma unroll
    for (int i = 0; i < KNN; ++i)
      topk[(size_t)(row0 + tid) * KNN + i] = fi[i];
  }
}

// ---------------------------------------------------------------------------
// Kernel 3: loss = 100 * sum_i (16 - 2*|setX_i ∩ setZ_i|) / n^2
// ---------------------------------------------------------------------------
__global__ void zero_out_kernel(float* out) { out[0] = 0.0f; }

__global__ void bce_kernel(const int* __restrict__ idxX,
                           const int* __restrict__ idxZ,
                           float* __restrict__ out) {
  int row = blockIdx.x * blockDim.x + threadIdx.x;
  if (row >= N_PTS) return;
  int a[KNN], b[KNN];
#pragma unroll
  for (int i = 0; i < KNN; ++i) {
    a[i] = idxX[(size_t)row * KNN + i];
    b[i] = idxZ[(size_t)row * KNN + i];
  }
  int ov = 0;
#pragma unroll
  for (int i = 0; i < KNN; ++i)
#pragma unroll
    for (int j = 0; j < KNN; ++j)
      ov += (a[i] == b[j]);
  float contrib = 100.0f * (float)(2 * KNN - 2 * ov) /
                  ((float)N_PTS * (float)N_PTS);
  atomicAdd(out, contrib);
}

// ---------------------------------------------------------------------------
extern "C" void kernel_launch(void* const* d_in, const int* in_sizes, int n_in,
                              void* d_out, int out_size, void* d_ws, size_t ws_size,
                              hipStream_t stream) {
  const float* X = (const float*)d_in[0];  // 8192 x 128
  const float* Z = (const float*)d_in[1];  // 8192 x 32
  float* out = (float*)d_out;

  // Workspace layout
  float* sqX  = (float*)d_ws;
  float* sqZ  = sqX + N_PTS;
  int*   idxX = (int*)(sqZ + N_PTS);
  int*   idxZ = idxX + (size_t)N_PTS * KNN;

  row_norms_kernel<<<(N_PTS + 127) / 128, 128, 0, stream>>>(X, 128, sqX);
  row_norms_kernel<<<(N_PTS + 127) / 128, 128, 0, stream>>>(Z, 32, sqZ);

  knn_topk_kernel<<<N_PTS / ROWS, THREADS, 0, stream>>>(X, 128, sqX, idxX);
  knn_topk_kernel<<<N_PTS / ROWS, THREADS, 0, stream>>>(Z, 32, sqZ, idxZ);

  zero_out_kernel<<<1, 1, 0, stream>>>(out);
  bce_kernel<<<(N_PTS + 255) / 256, 256, 0, stream>>>(idxX, idxZ, out);
}